// ConvCNP2D_87771951661186
// MI455X (gfx1250) — compile-verified
//
#include <hip/hip_runtime.h>

typedef __attribute__((ext_vector_type(2))) float v2f;
typedef __attribute__((ext_vector_type(8))) float v8f;

#define WMMA_F32_4(A, B, C) \
  __builtin_amdgcn_wmma_f32_16x16x4_f32(false, (A), false, (B), (short)0, (C), false, false)

#define EXP2F(x) __builtin_amdgcn_exp2f(x)   // raw v_exp_f32 (2^x)
#define LOG2E 1.4426950408889634f

// ---------------------------------------------------------------------------
// Stage 1: SetConv (context points -> grid), density-normalized, projected
// to 16 channels + sigmoid. Output layout act0[b][c][gy][gx] (NCHW with
// H=gy, W=gx) exactly matching the reference's transpose/swapaxes chain.
// exp realized as exp2 with log2(e) folded into the per-channel factor.
// ---------------------------------------------------------------------------
__global__ __launch_bounds__(256) void setconv_in_kernel(
    const float* __restrict__ x, const float* __restrict__ y,
    const float* __restrict__ sigma_in, const float* __restrict__ W_in,
    const float* __restrict__ b_in, float* __restrict__ act0)
{
  int tid = blockIdx.x * blockDim.x + threadIdx.x;   // 0..4095
  int b = tid >> 10;
  int m = tid & 1023;                                 // m = gx*32 + gy
  int gx = m >> 5, gy = m & 31;
  const float inv31 = 1.0f / 31.0f;
  float tmx = gx * inv31, tmy = gy * inv31;
  float f0 = -0.5f * LOG2E * __expf(-2.0f * sigma_in[0]);  // -0.5*log2e/exp(s)^2
  float f1 = -0.5f * LOG2E * __expf(-2.0f * sigma_in[1]);
  float acc0 = 0.0f, acc1 = 0.0f;
  const float* xb = x + b * 256 * 2;
  const float* yb = y + b * 256;
  for (int n = 0; n < 256; ++n) {
    float dx = xb[2 * n] - tmx, dy = xb[2 * n + 1] - tmy;
    float d = dx * dx + dy * dy;
    acc0 += EXP2F(d * f0);                            // density channel (y==1)
    acc1 += EXP2F(d * f1) * yb[n];
  }
  float feat0 = acc0;
  float feat1 = acc1 / (acc0 + 1e-8f);
  #pragma unroll
  for (int c = 0; c < 16; ++c) {
    float v = W_in[2 * c] * feat0 + W_in[2 * c + 1] * feat1 + b_in[c];
    float h = 1.0f / (1.0f + __expf(-v));             // sigmoid
    act0[((b * 16 + c) * 32 + gy) * 32 + gx] = h;
  }
}

// ---------------------------------------------------------------------------
// Stage 2: 5x5 SAME conv as implicit GEMM on V_WMMA_F32_16X16X4_F32.
// One wave per 16(spatial) x 16(cout) output tile; K = CIN*25 in steps of 4.
// Zero-padding realized branchlessly: clamp coordinates to a valid address,
// load unconditionally, multiply by 0/1 mask -> no exec-mask juggling around
// the WMMA chain, loads schedulable ahead of the MMA.
// ---------------------------------------------------------------------------
template <int CIN, int COUT, bool RELU, bool TOUT>
__global__ __launch_bounds__(32) void conv5x5_wmma_kernel(
    const float* __restrict__ in, const float* __restrict__ W,
    const float* __restrict__ bias, float* __restrict__ out)
{
  constexpr int NT = COUT / 16;
  int blk = blockIdx.x;
  int ntile = blk % NT;
  int stile = (blk / NT) % 64;
  int b = blk / (NT * 64);
  int lane = threadIdx.x;
  int half = lane >> 4;
  int lm = lane & 15;                 // A row (M) and B/C column (N)
  int gy0 = stile >> 1;               // 16 consecutive s share one gy row
  int gx0 = (stile & 1) * 16;
  int o = ntile * 16 + lm;            // output channel for B/C column
  int gxA = gx0 + lm;                 // gx of this lane's A row

  v8f acc = {};
  const float* inb = in + b * CIN * 1024;
  const int KDIM = CIN * 25;
  for (int kb = 0; kb < KDIM; kb += 4) {
    int k0 = kb + half * 2;           // K index for v2f component 0
    v2f av, bv;
    {
      int k = k0;
      int cin = k / 25; int r = k - cin * 25;
      int kh = r / 5;   int kw = r - kh * 5;
      int gyi = gy0 + kh - 2;
      int gxi = gxA + kw - 2;
      float msk = ((gyi >= 0) & (gyi < 32) & (gxi >= 0) & (gxi < 32)) ? 1.0f : 0.0f;
      int gyc = min(max(gyi, 0), 31);
      int gxc = min(max(gxi, 0), 31);
      av.x = inb[(cin * 32 + gyc) * 32 + gxc] * msk;
      bv.x = W[(o * CIN + cin) * 25 + r];
    }
    {
      int k = k0 + 1;
      int cin = k / 25; int r = k - cin * 25;
      int kh = r / 5;   int kw = r - kh * 5;
      int gyi = gy0 + kh - 2;
      int gxi = gxA + kw - 2;
      float msk = ((gyi >= 0) & (gyi < 32) & (gxi >= 0) & (gxi < 32)) ? 1.0f : 0.0f;
      int gyc = min(max(gyi, 0), 31);
      int gxc = min(max(gxi, 0), 31);
      av.y = inb[(cin * 32 + gyc) * 32 + gxc] * msk;
      bv.y = W[(o * CIN + cin) * 25 + r];
    }
    acc = WMMA_F32_4(av, bv, acc);
  }
  float bo = bias[o];
  #pragma unroll
  for (int v = 0; v < 8; ++v) {
    int M = v + half * 8;             // C layout: VGPR v, half selects M or M+8
    float val = acc[v] + bo;
    if (RELU) val = fmaxf(val, 0.0f);
    int gxo = gx0 + M;
    if (TOUT) {
      int mref = gxo * 32 + gy0;      // back to reference m = gx*32+gy
      out[(b * 1024 + mref) * 16 + o] = val;
    } else {
      out[((b * COUT + o) * 32 + gy0) * 32 + gxo] = val;
    }
  }
}

// ---------------------------------------------------------------------------
// Stage 3: grid->grid SetConv for mean & var heads as one WMMA GEMM chain.
// K = (n, c) = 1024*16; A[m][k] = exp(-0.5*d(m,n)/s_c^2) * W_head[c] built
// on the fly (grid coords derived arithmetically), B[k][col] = h[b=col][n][c].
// Columns 4..15 of C accumulate duplicated batch-0 data but are never
// stored (WMMA columns are independent), so no masking is needed at all.
// Two accumulators (mean/var) share the B fragment. K-loop split over NW
// waves; partials reduced through LDS with the workgroup split-barrier.
// ---------------------------------------------------------------------------
#define NW 8                          // waves per workgroup (n-slices of 128)

__global__ __launch_bounds__(32 * NW) void setconv_out_kernel(
    const float* __restrict__ h3,
    const float* __restrict__ sigma_m, const float* __restrict__ W_m,
    const float* __restrict__ b_m,
    const float* __restrict__ sigma_s, const float* __restrict__ W_s,
    const float* __restrict__ b_s,
    float* __restrict__ out)
{
  __shared__ float red[NW][16][32];   // [wave][head*8+v][lane], conflict-free

  int mt = blockIdx.x;                // 0..63 target-row tile
  int wave = threadIdx.x >> 5;
  int lane = threadIdx.x & 31;
  int half = lane >> 4;
  int col = lane & 15;                // A row (M) and B/C column (N)
  int mrow = mt * 16 + col;
  int gxm = mrow >> 5, gym = mrow & 31;
  const float inv31 = 1.0f / 31.0f;

  // This lane only ever touches channels c = 4*q + 2*half + j, q=0..3, j=0..1.
  // log2(e) folded into the factors: exp(d*f) == exp2(d*(f*log2e)).
  float fm0[4], fm1[4], fs0[4], fs1[4];
  float wm0[4], wm1[4], ws0[4], ws1[4];
  #pragma unroll
  for (int q = 0; q < 4; ++q) {
    int c0 = 4 * q + half * 2;
    fm0[q] = -0.5f * LOG2E * __expf(-2.0f * sigma_m[c0]);     wm0[q] = W_m[c0];
    fm1[q] = -0.5f * LOG2E * __expf(-2.0f * sigma_m[c0 + 1]); wm1[q] = W_m[c0 + 1];
    fs0[q] = -0.5f * LOG2E * __expf(-2.0f * sigma_s[c0]);     ws0[q] = W_s[c0];
    fs1[q] = -0.5f * LOG2E * __expf(-2.0f * sigma_s[c0 + 1]); ws1[q] = W_s[c0 + 1];
  }

  const float* hb = h3 + (col < 4 ? col : 0) * 1024 * 16;  // valid for all lanes

  v8f accM = {}, accS = {};
  // this wave's n-slice: gxn in [4*wave, 4*wave+4)
  for (int gxn = 4 * wave; gxn < 4 * wave + 4; ++gxn) {
    float dx = (float)(gxm - gxn) * inv31;
    float dx2 = dx * dx;
    const float* hrow = hb + gxn * 32 * 16;
    __builtin_prefetch(hrow + 32 * 16, 0, 3);  // next gxn block of h stream
    for (int gyn = 0; gyn < 32; ++gyn) {
      float dy = (float)(gym - gyn) * inv31;
      float d = dx2 + dy * dy;        // shared across the 4 K-steps of this n
      const float* hn = hrow + gyn * 16;
      #pragma unroll
      for (int q = 0; q < 4; ++q) {
        int c0 = 4 * q + half * 2;    // even -> 8B-aligned pair load
        v2f bb = *(const v2f*)(hn + c0);   // unmasked: dead columns never stored
        v2f am, as;
        am.x = EXP2F(d * fm0[q]) * wm0[q];
        am.y = EXP2F(d * fm1[q]) * wm1[q];
        as.x = EXP2F(d * fs0[q]) * ws0[q];
        as.y = EXP2F(d * fs1[q]) * ws1[q];
        accM = WMMA_F32_4(am, bb, accM);
        accS = WMMA_F32_4(as, bb, accS);
      }
    }
  }

  // cross-wave reduction of the two C fragments through LDS
  #pragma unroll
  for (int v = 0; v < 8; ++v) {
    red[wave][v][lane]     = accM[v];
    red[wave][8 + v][lane] = accS[v];
  }
  __syncthreads();

  if (wave == 0) {
    float sm[8], ss[8];
    #pragma unroll
    for (int v = 0; v < 8; ++v) { sm[v] = 0.0f; ss[v] = 0.0f; }
    #pragma unroll
    for (int w = 0; w < NW; ++w) {
      #pragma unroll
      for (int v = 0; v < 8; ++v) {
        sm[v] += red[w][v][lane];
        ss[v] += red[w][8 + v][lane];
      }
    }
    if (col < 4) {                    // column col == batch index
      int b = col;
      float bm = b_m[0], bs = b_s[0];
      #pragma unroll
      for (int v = 0; v < 8; ++v) {
        int m = mt * 16 + v + half * 8;
        out[b * 2048 + m]        = sm[v] + bm;   // means block
        out[b * 2048 + 1024 + m] = ss[v] + bs;   // var block
      }
    }
  }
}

// ---------------------------------------------------------------------------
extern "C" void kernel_launch(void* const* d_in, const int* in_sizes, int n_in,
                              void* d_out, int out_size, void* d_ws, size_t ws_size,
                              hipStream_t stream)
{
  const float* x        = (const float*)d_in[0];
  const float* y        = (const float*)d_in[1];
  // d_in[2] = x_grid: derived analytically (linspace(0,1,32)), not needed
  const float* sigma_in = (const float*)d_in[3];
  const float* W_in     = (const float*)d_in[4];
  const float* b_in     = (const float*)d_in[5];
  const float* sigma_m  = (const float*)d_in[6];
  const float* W_m      = (const float*)d_in[7];
  const float* b_m      = (const float*)d_in[8];
  const float* sigma_s  = (const float*)d_in[9];
  const float* W_s      = (const float*)d_in[10];
  const float* b_s      = (const float*)d_in[11];
  const float* cW1      = (const float*)d_in[12];
  const float* cb1      = (const float*)d_in[13];
  const float* cW2      = (const float*)d_in[14];
  const float* cb2      = (const float*)d_in[15];
  const float* cW3      = (const float*)d_in[16];
  const float* cb3      = (const float*)d_in[17];
  float* outp = (float*)d_out;

  float* act0 = (float*)d_ws;                 // [4][16][32][32]
  float* act1 = act0 + 4 * 16 * 1024;         // [4][32][32][32]
  float* act2 = act1 + 4 * 32 * 1024;         // [4][32][32][32]
  float* h3   = act2 + 4 * 32 * 1024;         // [4][1024][16]

  setconv_in_kernel<<<16, 256, 0, stream>>>(x, y, sigma_in, W_in, b_in, act0);
  conv5x5_wmma_kernel<16, 32, true,  false><<<4 * 64 * 2, 32, 0, stream>>>(act0, cW1, cb1, act1);
  conv5x5_wmma_kernel<32, 32, true,  false><<<4 * 64 * 2, 32, 0, stream>>>(act1, cW2, cb2, act2);
  conv5x5_wmma_kernel<32, 16, false, true ><<<4 * 64 * 1, 32, 0, stream>>>(act2, cW3, cb3, h3);
  setconv_out_kernel<<<64, 32 * NW, 0, stream>>>(h3, sigma_m, W_m, b_m,
                                                 sigma_s, W_s, b_s, outp);
}